// FCGAT_27556510171605
// MI455X (gfx1250) — compile-verified
//
#include <hip/hip_runtime.h>

typedef float v2f __attribute__((ext_vector_type(2)));
typedef float v8f __attribute__((ext_vector_type(8)));

constexpr int Nb   = 64;
constexpr int Kn   = 1024;
constexpr int D    = 128;
constexpr int Mtot = Nb * Kn;           // 65536 rows of x / h
constexpr int STRIPS_PER_BLOCK = 8;

// ---------------------------------------------------------------------------
// prep: u1[d] = sum_e W[d,e]*a_w[e], u2[d] = sum_e W[d,e]*a_w[128+e]
//       C = bW.a1 + bW.a2 + a_b   (all score-bias terms folded together)
// ---------------------------------------------------------------------------
__global__ __launch_bounds__(128) void prep_kernel(const float* __restrict__ W,
                                                   const float* __restrict__ bW,
                                                   const float* __restrict__ a_w,
                                                   const float* __restrict__ a_b,
                                                   float* __restrict__ u,
                                                   float* __restrict__ Cp) {
  const int d = threadIdx.x;
  float acc1 = 0.f, acc2 = 0.f;
  for (int e = 0; e < D; ++e) {
    const float w = W[d * D + e];
    acc1 += w * a_w[e];
    acc2 += w * a_w[D + e];
  }
  u[d]     = acc1;
  u[D + d] = acc2;
  if (d == 0) {
    float c = a_b[0];
    for (int e = 0; e < D; ++e) c += bW[e] * (a_w[e] + a_w[D + e]);
    Cp[0] = c;
  }
}

// ---------------------------------------------------------------------------
// sdots: sI[m] = x[m,:].u1 ; sJ[m] = x[m,:].u2   (one wave per row)
// ---------------------------------------------------------------------------
__global__ __launch_bounds__(256) void sdots_kernel(const float* __restrict__ x,
                                                    const float* __restrict__ u,
                                                    float* __restrict__ sI,
                                                    float* __restrict__ sJ) {
  const int wave = threadIdx.x >> 5;
  const int lane = threadIdx.x & 31;
  const int m = blockIdx.x * 8 + wave;
  const float4 xa = *(const float4*)(x + (size_t)m * D + lane * 4);
  const float4 u1 = *(const float4*)(u + lane * 4);
  const float4 u2 = *(const float4*)(u + D + lane * 4);
  float a1 = xa.x * u1.x + xa.y * u1.y + xa.z * u1.z + xa.w * u1.w;
  float a2 = xa.x * u2.x + xa.y * u2.y + xa.z * u2.z + xa.w * u2.w;
  for (int o = 16; o > 0; o >>= 1) {
    a1 += __shfl_xor(a1, o, 32);
    a2 += __shfl_xor(a2, o, 32);
  }
  if (lane == 0) { sI[m] = a1; sJ[m] = a2; }
}

// ---------------------------------------------------------------------------
// factor[m] = sum_j softmax_j(lrelu(sI[m]+sJ[n,j]+C))   (one wave per row)
// 3 faithful passes: max, Z, sum(e/Z). sJ row staged in LDS per block.
// ---------------------------------------------------------------------------
__global__ __launch_bounds__(256) void factor_kernel(const float* __restrict__ sI,
                                                     const float* __restrict__ sJ,
                                                     const float* __restrict__ Cp,
                                                     float* __restrict__ fac) {
  __shared__ float sj[Kn];
  const int tid = threadIdx.x;
  const int m0 = blockIdx.x * 8;
  const int base = (m0 >> 10) << 10;       // all 8 rows share graph index n
  *(float4*)(sj + tid * 4) = *(const float4*)(sJ + base + tid * 4);
  __syncthreads();

  const int wave = tid >> 5, lane = tid & 31;
  const int m = m0 + wave;
  const float si = sI[m] + Cp[0];

  float mx = -3.4e38f;
#pragma unroll 4
  for (int it = 0; it < 32; ++it) {
    float v = si + sj[lane + 32 * it];
    v = fmaxf(v, 0.01f * v);               // LeakyReLU(0.01)
    mx = fmaxf(mx, v);
  }
  for (int o = 16; o > 0; o >>= 1) mx = fmaxf(mx, __shfl_xor(mx, o, 32));

  float Z = 0.f;
#pragma unroll 4
  for (int it = 0; it < 32; ++it) {
    float v = si + sj[lane + 32 * it];
    v = fmaxf(v, 0.01f * v);
    Z += __expf(v - mx);
  }
  for (int o = 16; o > 0; o >>= 1) Z += __shfl_xor(Z, o, 32);
  const float rz = 1.f / Z;

  float F = 0.f;
#pragma unroll 4
  for (int it = 0; it < 32; ++it) {
    float v = si + sj[lane + 32 * it];
    v = fmaxf(v, 0.01f * v);
    F += __expf(v - mx) * rz;
  }
  for (int o = 16; o > 0; o >>= 1) F += __shfl_xor(F, o, 32);
  if (lane == 0) fac[m] = F;
}

// ---------------------------------------------------------------------------
// gat_gemm: out[m,n] = relu((x[m,:] @ W[:,n] + bW[n]) * fac[m])
// fp32 WMMA 16x16x4. Wave w owns columns 16w..16w+15; B fragments for the
// full K=128 live in 32 v2f registers. A strips staged LDS -> registers so
// all 16 ds_load_2addr_b64 issue before a single dscnt wait, then 32 WMMAs
// run back-to-back.
// ---------------------------------------------------------------------------
__global__ __launch_bounds__(256) void gat_gemm_kernel(const float* __restrict__ x,
                                                       const float* __restrict__ W,
                                                       const float* __restrict__ bW,
                                                       const float* __restrict__ fac,
                                                       float* __restrict__ out) {
  __shared__ float sA[16 * 132];           // 8448 B, conflict-free A fragments
  const int tid  = threadIdx.x;
  const int wave = tid >> 5;
  const int lane = tid & 31;
  const int hl   = lane >> 4;              // 0: K {0,1}, 1: K {2,3}
  const int ln   = lane & 15;
  const int ncol = wave * 16 + ln;         // output column of this lane

  // B fragments (V_WMMA layout: VGPR0={K0,K2}, VGPR1={K1,K3} across halves)
  v2f breg[32];
#pragma unroll
  for (int s = 0; s < 32; ++s) {
    const int k = 4 * s + 2 * hl;
    v2f b;
    b.x = W[k * D + ncol];
    b.y = W[(k + 1) * D + ncol];
    breg[s] = b;
  }
  const float bias = bW[ncol];

  const int arow = tid >> 4;               // staging: row 0..15
  const int akk  = (tid & 15) * 8;         // staging: 8 consecutive K each

  for (int s = 0; s < STRIPS_PER_BLOCK; ++s) {
    const int m0 = (blockIdx.x * STRIPS_PER_BLOCK + s) * 16;

    const float* src = x + (size_t)(m0 + arow) * D + akk;
    float* dst = sA + arow * 132 + akk;
    *(float4*)(dst)     = *(const float4*)(src);
    *(float4*)(dst + 4) = *(const float4*)(src + 4);
    __syncthreads();

    // hoist factor loads so they're in flight during the WMMA chain
    float fr[8];
#pragma unroll
    for (int r = 0; r < 8; ++r) fr[r] = fac[m0 + r + 8 * hl];

    // stage the whole A strip for this lane into registers first
    const float* ap = sA + ln * 132 + 2 * hl;
    v2f afrag[32];
#pragma unroll
    for (int sk = 0; sk < 32; ++sk) {
      afrag[sk] = *(const v2f*)(ap + 4 * sk);
    }

    v8f acc;
#pragma unroll
    for (int r = 0; r < 8; ++r) acc[r] = bias;

#pragma unroll
    for (int sk = 0; sk < 32; ++sk) {
      acc = __builtin_amdgcn_wmma_f32_16x16x4_f32(false, afrag[sk], false, breg[sk],
                                                  (short)0, acc, false, false);
    }

#pragma unroll
    for (int r = 0; r < 8; ++r) {
      const int m = m0 + r + 8 * hl;
      const float v = acc[r] * fr[r];
      out[(size_t)m * D + ncol] = fmaxf(v, 0.f);
    }
    __syncthreads();
  }
}

// ---------------------------------------------------------------------------
extern "C" void kernel_launch(void* const* d_in, const int* in_sizes, int n_in,
                              void* d_out, int out_size, void* d_ws, size_t ws_size,
                              hipStream_t stream) {
  const float* x   = (const float*)d_in[0];
  const float* W   = (const float*)d_in[1];
  const float* bW  = (const float*)d_in[2];
  const float* a_w = (const float*)d_in[3];
  const float* a_b = (const float*)d_in[4];
  float* out = (float*)d_out;

  float* ws  = (float*)d_ws;
  float* u   = ws;                  // 256 floats: u1 | u2
  float* Cp  = ws + 256;            // 1 float
  float* sI  = ws + 1024;           // 65536 floats
  float* sJ  = ws + 1024 + Mtot;    // 65536 floats
  float* fac = ws + 1024 + 2 * Mtot;// 65536 floats  (~0.79 MB total)

  prep_kernel<<<1, 128, 0, stream>>>(W, bW, a_w, a_b, u, Cp);
  sdots_kernel<<<Mtot / 8, 256, 0, stream>>>(x, u, sI, sJ);
  factor_kernel<<<Mtot / 8, 256, 0, stream>>>(sI, sJ, Cp, fac);
  gat_gemm_kernel<<<Mtot / (16 * STRIPS_PER_BLOCK), 256, 0, stream>>>(x, W, bW, fac, out);
}